// CELoss_28656021799139
// MI455X (gfx1250) — compile-verified
//
#include <hip/hip_runtime.h>
#include <hip/hip_bf16.h>

// CE loss, class-balanced mean. B=4, C=8, D=64, H=128, W=128.
// DHW = 2^20, N = 4*DHW = 4,194,304. Bandwidth-bound streaming reduction.

typedef __attribute__((ext_vector_type(2))) float v2f;
typedef __attribute__((ext_vector_type(8))) float v8f;

static constexpr int      CH   = 8;
static constexpr unsigned DHW  = 1u << 20;      // 64*128*128
static constexpr unsigned NTOT = 4u << 20;      // B * DHW

// ---------------------------------------------------------------------------
// Kernel 1: streaming logsumexp + per-class scatter into registers.
// Each thread handles 4 consecutive spatial elements per iteration via
// float4/int4 loads (all 8 channel streams coalesced, 4 MiB apart).
// Wave32 butterfly reduce -> per-block 16 partials (8 sums, 8 counts) in ws.
// ---------------------------------------------------------------------------
__global__ __launch_bounds__(256) void ce_partial(const float* __restrict__ logits,
                                                  const int*   __restrict__ labels,
                                                  float*       __restrict__ ws) {
  float accS[CH], accN[CH];
#pragma unroll
  for (int c = 0; c < CH; ++c) { accS[c] = 0.0f; accN[c] = 0.0f; }

  const unsigned tid  = blockIdx.x * blockDim.x + threadIdx.x;
  const unsigned step = gridDim.x * blockDim.x * 4u;   // elements per grid pass

  for (unsigned base = tid * 4u; base < NTOT; base += step) {
    const unsigned b = base >> 20;            // batch index
    const unsigned r = base & (DHW - 1u);     // offset within [D,H,W]
    const float* p = logits + (((size_t)b * CH) << 20) + r;

    float x[CH][4];
#pragma unroll
    for (int c = 0; c < CH; ++c) {
      const float4 q = *reinterpret_cast<const float4*>(p + ((size_t)c << 20));
      x[c][0] = q.x; x[c][1] = q.y; x[c][2] = q.z; x[c][3] = q.w;
      // gfx1250 global_prefetch_b8: pull next grid-stride tile toward L2/L0.
      __builtin_prefetch(p + ((size_t)c << 20) + step, 0, 3);
    }
    const int4 lq = *reinterpret_cast<const int4*>(labels + base);
    __builtin_prefetch(labels + base + step, 0, 3);
    const int lab[4] = {lq.x, lq.y, lq.z, lq.w};

#pragma unroll
    for (int j = 0; j < 4; ++j) {
      float m = x[0][j];
#pragma unroll
      for (int c = 1; c < CH; ++c) m = fmaxf(m, x[c][j]);
      float s = 0.0f;
#pragma unroll
      for (int c = 0; c < CH; ++c) s += __expf(x[c][j] - m);
      const float lse = m + __logf(s);
#pragma unroll
      for (int c = 0; c < CH; ++c) {           // branchless one-hot scatter
        const float f = (lab[j] == c) ? 1.0f : 0.0f;
        accS[c] = fmaf(f, lse - x[c][j], accS[c]);
        accN[c] += f;
      }
    }
  }

  // wave32 butterfly reduction (5 steps, 32 lanes)
#pragma unroll
  for (int mask = 16; mask >= 1; mask >>= 1) {
#pragma unroll
    for (int c = 0; c < CH; ++c) {
      accS[c] += __shfl_xor(accS[c], mask, 32);
      accN[c] += __shfl_xor(accN[c], mask, 32);
    }
  }

  __shared__ float red[8][16];                 // 8 waves @ blockDim 256
  const unsigned lane = threadIdx.x & 31u;
  const unsigned wid  = threadIdx.x >> 5;
  if (lane == 0) {
#pragma unroll
    for (int c = 0; c < CH; ++c) { red[wid][c] = accS[c]; red[wid][CH + c] = accN[c]; }
  }
  __syncthreads();
  if (threadIdx.x < 16) {
    float v = 0.0f;
#pragma unroll
    for (int w = 0; w < 8; ++w) v += red[w][threadIdx.x];
    ws[(size_t)blockIdx.x * 16 + threadIdx.x] = v;   // [block][0..7]=sums, [8..15]=counts
  }
}

// ---------------------------------------------------------------------------
// Kernel 2 (single wave32): reduce nrows x 16 partials -> 16 totals via
// V_WMMA_F32_16X16X4_F32 with a ones A-matrix (exact f32 column sums:
// D[m,n] = sum_k B[k,n] + C[m,n]). Two independent accumulator chains hide
// the WMMA->WMMA accumulation latency. Then class-balanced mean.
// A 16x4 f32 layout (ISA 7.12.2): VGPR0 = K0|K2, VGPR1 = K1|K3 -> ones.
// B 4x16 f32 layout (symmetric): lanes 0-15 carry N, VGPR0=K0|K2, VGPR1=K1|K3.
// D: acc[0], lanes 0-15 = row M=0, columns N=0..15.
// ---------------------------------------------------------------------------
__global__ __launch_bounds__(32) void ce_final(const float* __restrict__ ws,
                                               float*       __restrict__ out,
                                               int nrows) {                 // multiple of 8
  const int lane = threadIdx.x;
  const int k0   = (lane < 16) ? 0 : 2;
  const int col  = lane & 15;

  const v2f ones = {1.0f, 1.0f};
  v8f acc0 = {};
  v8f acc1 = {};

  const int ngroups = nrows >> 2;              // 4 rows per WMMA, even count
  for (int g = 0; g < ngroups; g += 2) {
    const int r0 = (g << 2) + k0;
    const int r1 = ((g + 1) << 2) + k0;
    v2f b0, b1;
    b0[0] = ws[r0 * 16 + col];
    b0[1] = ws[(r0 + 1) * 16 + col];
    b1[0] = ws[r1 * 16 + col];
    b1[1] = ws[(r1 + 1) * 16 + col];
    acc0 = __builtin_amdgcn_wmma_f32_16x16x4_f32(false, ones, false, b0,
                                                 (short)0, acc0, false, false);
    acc1 = __builtin_amdgcn_wmma_f32_16x16x4_f32(false, ones, false, b1,
                                                 (short)0, acc1, false, false);
  }
  const v8f acc = acc0 + acc1;

  const float tot = acc[0];                               // lane n<16: total[n]
  const float cnt = __shfl(tot, (lane + 8) & 31, 32);     // lane c<8: count[c]
  float mean = 0.0f, pres = 0.0f;
  if (lane < 8 && cnt > 0.0f) { mean = tot / cnt; pres = 1.0f; }
#pragma unroll
  for (int mask = 4; mask >= 1; mask >>= 1) {             // reduce lanes 0..7
    mean += __shfl_xor(mean, mask, 32);
    pres += __shfl_xor(pres, mask, 32);
  }
  if (lane == 0) out[0] = mean / fmaxf(pres, 1.0f);
}

// ---------------------------------------------------------------------------
extern "C" void kernel_launch(void* const* d_in, const int* in_sizes, int n_in,
                              void* d_out, int out_size, void* d_ws, size_t ws_size,
                              hipStream_t stream) {
  (void)in_sizes; (void)n_in; (void)out_size;
  const float* logits = (const float*)d_in[0];
  const int*   labels = (const int*)d_in[1];
  float*       out    = (float*)d_out;
  float*       ws     = (float*)d_ws;

  int nblk = 1024;                                        // 64 KB of ws partials
  const size_t maxblk = ws_size / (16 * sizeof(float));
  if ((size_t)nblk > maxblk) nblk = (int)(maxblk & ~(size_t)7);
  if (nblk < 8) nblk = 8;

  ce_partial<<<nblk, 256, 0, stream>>>(logits, labels, ws);
  ce_final<<<1, 32, 0, stream>>>(ws, out, nblk);
}